// GAT_73839077753146
// MI455X (gfx1250) — compile-verified
//
#include <hip/hip_runtime.h>
#include <math.h>

typedef __attribute__((ext_vector_type(2))) float v2f;
typedef __attribute__((ext_vector_type(8))) float v8f;

#define EMB  128
#define HC1  256
#define H1   4
#define C1   64
#define OUTC 64

// ---------------------------------------------------------------------------
// Ordered-key mapping so float max can use integer atomicMax.
// encode: monotone bijection float -> uint.  Every dst node has a self-loop,
// so an init key of 0 (below every valid float key) is always overwritten.
// ---------------------------------------------------------------------------
__device__ __forceinline__ unsigned key_encode(float f) {
  unsigned u = __float_as_uint(f);
  return (u & 0x80000000u) ? ~u : (u | 0x80000000u);
}
__device__ __forceinline__ float key_decode(unsigned k) {
  unsigned u = (k & 0x80000000u) ? (k & 0x7fffffffu) : ~k;
  return __uint_as_float(u);
}

// ---------------------------------------------------------------------------
// C[M,Nc] = A[M,K] @ B[Nc,K]^T via V_WMMA_F32_16X16X4_F32 (full f32 precision).
// Each wave computes a 16(M) x 64(N) strip: 4 accumulator fragments share one
// A fragment per k-step.  Software-pipelined by one k-step: fragments for
// k+4 are loaded before the 4 WMMAs consume step k, so loads clause-batch and
// overlap the XDL pipe instead of load->wait->wmma lockstep.
// Fragment layouts per CDNA5 ISA 7.12.2:
//   A 16x4 : lane L holds M=L%16, K = 2*(L/16)+v   (v = frag element 0/1)
//   B 4x16 : lane L holds N=L%16, K = 2*(L/16)+v
//   C 16x16: frag r, lane L -> M = r + 8*(L/16), N = L%16
// ---------------------------------------------------------------------------
__global__ void gemm_wmma_f32_x4(const float* __restrict__ A,
                                 const float* __restrict__ B,
                                 float* __restrict__ C,
                                 int M, int K, int Nc) {
  const int lane = threadIdx.x & 31;
  const int wave = threadIdx.x >> 5;
  const int col_groups = Nc >> 6;                 // 64-wide strips
  const int total = (M >> 4) * col_groups;
  const int tile = blockIdx.x * (blockDim.x >> 5) + wave;
  if (tile >= total) return;
  const int tm = tile / col_groups;
  const int cg = tile - tm * col_groups;
  const int row0 = tm << 4, col0 = cg << 6;
  const int mr = lane & 15, half = lane >> 4;

  const float* Ap = A + (size_t)(row0 + mr) * K + 2 * half;
  const float* Bp = B + (size_t)(col0 + mr) * K + 2 * half;
  const size_t bs = (size_t)16 * K;               // next 16-col tile of W

  v8f acc0 = {}, acc1 = {}, acc2 = {}, acc3 = {};

  // prologue: load k-step 0
  v2f a  = *(const v2f*)(Ap);
  v2f b0 = *(const v2f*)(Bp);
  v2f b1 = *(const v2f*)(Bp + bs);
  v2f b2 = *(const v2f*)(Bp + 2 * bs);
  v2f b3 = *(const v2f*)(Bp + 3 * bs);

  for (int k0 = 0; k0 < K; k0 += 4) {
    // preload next k-step (clamped: last iteration re-loads a valid address,
    // values unused) so loads overlap the WMMA chain below
    const int k1 = (k0 + 4 < K) ? (k0 + 4) : k0;
    v2f an  = *(const v2f*)(Ap + k1);
    v2f b0n = *(const v2f*)(Bp + k1);
    v2f b1n = *(const v2f*)(Bp + bs + k1);
    v2f b2n = *(const v2f*)(Bp + 2 * bs + k1);
    v2f b3n = *(const v2f*)(Bp + 3 * bs + k1);

    acc0 = __builtin_amdgcn_wmma_f32_16x16x4_f32(false, a, false, b0, (short)0, acc0, false, false);
    acc1 = __builtin_amdgcn_wmma_f32_16x16x4_f32(false, a, false, b1, (short)0, acc1, false, false);
    acc2 = __builtin_amdgcn_wmma_f32_16x16x4_f32(false, a, false, b2, (short)0, acc2, false, false);
    acc3 = __builtin_amdgcn_wmma_f32_16x16x4_f32(false, a, false, b3, (short)0, acc3, false, false);

    a = an; b0 = b0n; b1 = b1n; b2 = b2n; b3 = b3n;
  }

  float* Cp = C + (size_t)(row0 + 8 * half) * Nc + col0 + mr;
#pragma unroll
  for (int r = 0; r < 8; ++r) {
    float* row = Cp + (size_t)r * Nc;
    row[0]  = acc0[r];
    row[16] = acc1[r];
    row[32] = acc2[r];
    row[48] = acc3[r];
  }
}

// ---------------------------------------------------------------------------
// e_src[n,h] = sum_c h[n,h*64+c]*att_src[h,c]  (and same for e_dst).
// One wave per (node,head); each lane loads a float2, shfl-xor reduce.
// ---------------------------------------------------------------------------
template <int H>
__global__ void escore_kernel(const float* __restrict__ hfeat,
                              const float* __restrict__ asrc,
                              const float* __restrict__ adst,
                              float* __restrict__ e_src,
                              float* __restrict__ e_dst, int n) {
  const int lane = threadIdx.x & 31;
  const int wave = threadIdx.x >> 5;
  const int p = blockIdx.x * (blockDim.x >> 5) + wave;
  if (p >= n * H) return;
  const int node = p / H, h = p - node * H;      // H power of two -> shifts
  const int c = 2 * lane;
  const float* hp = hfeat + (size_t)node * (H * C1) + h * C1 + c;
  const float* as = asrc + h * C1 + c;
  const float* ad = adst + h * C1 + c;
  float s = hp[0] * as[0] + hp[1] * as[1];
  float d = hp[0] * ad[0] + hp[1] * ad[1];
#pragma unroll
  for (int m = 16; m; m >>= 1) {
    s += __shfl_xor(s, m, 32);
    d += __shfl_xor(d, m, 32);
  }
  if (lane == 0) { e_src[p] = s; e_dst[p] = d; }
}

__device__ __forceinline__ void edge_sd(const int* __restrict__ src,
                                        const int* __restrict__ dst,
                                        int e, int E, int& s, int& d) {
  if (e < E) { s = src[e]; d = dst[e]; }
  else       { s = e - E; d = s; }               // self-loops appended
}

// Pass 1: segment max of leaky_relu(e_src[s]+e_dst[d]) over incoming edges.
template <int H>
__global__ void edge_max_kernel(const int* __restrict__ src,
                                const int* __restrict__ dst,
                                const float* __restrict__ e_src,
                                const float* __restrict__ e_dst,
                                unsigned* __restrict__ mkey, int E, int n) {
  const int idx = blockIdx.x * blockDim.x + threadIdx.x;
  if (idx >= (E + n) * H) return;
  const int e = idx / H, h = idx & (H - 1);
  int s, d; edge_sd(src, dst, e, E, s, d);
  float v = e_src[s * H + h] + e_dst[d * H + h];
  v = v > 0.f ? v : 0.2f * v;
  atomicMax(&mkey[d * H + h], key_encode(v));
}

// Pass 2: w = exp(e - max); cache numerator, accumulate denominator.
template <int H>
__global__ void edge_denom_kernel(const int* __restrict__ src,
                                  const int* __restrict__ dst,
                                  const float* __restrict__ e_src,
                                  const float* __restrict__ e_dst,
                                  const unsigned* __restrict__ mkey,
                                  float* __restrict__ denom,
                                  float* __restrict__ wexp, int E, int n) {
  const int idx = blockIdx.x * blockDim.x + threadIdx.x;
  if (idx >= (E + n) * H) return;
  const int e = idx / H, h = idx & (H - 1);
  int s, d; edge_sd(src, dst, e, E, s, d);
  float v = e_src[s * H + h] + e_dst[d * H + h];
  v = v > 0.f ? v : 0.2f * v;
  const float w = __expf(v - key_decode(mkey[d * H + h]));
  wexp[idx] = w;
  atomicAdd(&denom[d * H + h], w);
}

// Pass 3: out[dst] += alpha * h[src].  Thread = (edge, 4-channel group);
// consecutive lanes cover consecutive channels -> coalesced gather/atomics.
template <int H>
__global__ void edge_agg_kernel(const int* __restrict__ src,
                                const int* __restrict__ dst,
                                const float* __restrict__ wexp,
                                const float* __restrict__ denom,
                                const float* __restrict__ hfeat,
                                float* __restrict__ out, int E, int n) {
  constexpr int HC = H * C1;
  constexpr int G = HC >> 2;                     // float4 groups per edge
  const long long idx = (long long)blockIdx.x * blockDim.x + threadIdx.x;
  if (idx >= (long long)(E + n) * G) return;
  const int e = (int)(idx >> (H == 4 ? 6 : 4));  // idx / G (G = 64 or 16)
  const int g = (int)idx & (G - 1);
  const int c0 = g << 2;
  const int h = c0 >> 6;                         // c0 / 64
  int s, d; edge_sd(src, dst, e, E, s, d);
  const float alpha = wexp[e * H + h] / (denom[d * H + h] + 1e-16f);
  const float4 hv = *(const float4*)(hfeat + (size_t)s * HC + c0);
  float* op = out + (size_t)d * HC + c0;
  atomicAdd(op + 0, alpha * hv.x);
  atomicAdd(op + 1, alpha * hv.y);
  atomicAdd(op + 2, alpha * hv.z);
  atomicAdd(op + 3, alpha * hv.w);
}

__global__ void bias_relu_kernel(float* __restrict__ data,
                                 const float* __restrict__ b, int total) {
  const int i = blockIdx.x * blockDim.x + threadIdx.x;
  if (i >= total) return;
  const float v = data[i] + b[i & (HC1 - 1)];
  data[i] = v > 0.f ? v : 0.f;
}

// bias + row-wise L2 normalize (64 channels, one wave per node)
__global__ void bias_norm_kernel(const float* __restrict__ in,
                                 const float* __restrict__ b,
                                 float* __restrict__ out, int n) {
  const int lane = threadIdx.x & 31;
  const int wave = threadIdx.x >> 5;
  const int node = blockIdx.x * (blockDim.x >> 5) + wave;
  if (node >= n) return;
  const int c = 2 * lane;
  const float v0 = in[(size_t)node * OUTC + c]     + b[c];
  const float v1 = in[(size_t)node * OUTC + c + 1] + b[c + 1];
  float ss = v0 * v0 + v1 * v1;
#pragma unroll
  for (int m = 16; m; m >>= 1) ss += __shfl_xor(ss, m, 32);
  const float scale = 1.0f / fmaxf(sqrtf(ss), 1e-12f);
  out[(size_t)node * OUTC + c]     = v0 * scale;
  out[(size_t)node * OUTC + c + 1] = v1 * scale;
}

// ---------------------------------------------------------------------------
extern "C" void kernel_launch(void* const* d_in, const int* in_sizes, int n_in,
                              void* d_out, int out_size, void* d_ws, size_t ws_size,
                              hipStream_t stream) {
  const float* x   = (const float*)d_in[0];
  const float* W1  = (const float*)d_in[1];
  const float* as1 = (const float*)d_in[2];
  const float* ad1 = (const float*)d_in[3];
  const float* b1  = (const float*)d_in[4];
  const float* W2  = (const float*)d_in[5];
  const float* as2 = (const float*)d_in[6];
  const float* ad2 = (const float*)d_in[7];
  const float* b2  = (const float*)d_in[8];
  const int*   ei  = (const int*)d_in[9];

  const int n = in_sizes[0] / EMB;       // 100000
  const int E = in_sizes[9] / 2;         // 1600000
  const int* srcI = ei;
  const int* dstI = ei + E;

  float* ws = (float*)d_ws;
  size_t off = 0;
  float*    hfeat1 = ws + off; off += (size_t)n * HC1;
  float*    out1   = ws + off; off += (size_t)n * HC1;   // reused as h1 in place
  float*    hfeat2 = ws + off; off += (size_t)n * OUTC;
  float*    out2   = ws + off; off += (size_t)n * OUTC;
  float*    esrc1  = ws + off; off += (size_t)n * H1;
  float*    edst1  = ws + off; off += (size_t)n * H1;
  unsigned* m1     = (unsigned*)(ws + off); off += (size_t)n * H1;
  float*    den1   = ws + off; off += (size_t)n * H1;
  float*    esrc2  = ws + off; off += (size_t)n;
  float*    edst2  = ws + off; off += (size_t)n;
  unsigned* m2     = (unsigned*)(ws + off); off += (size_t)n;
  float*    den2   = ws + off; off += (size_t)n;
  float*    wexp   = ws + off; off += (size_t)(E + n) * H1;  // reused by conv2

  hipMemsetAsync(out1, 0, (size_t)n * HC1 * 4, stream);
  hipMemsetAsync(out2, 0, (size_t)n * OUTC * 4, stream);
  hipMemsetAsync(m1,   0, (size_t)n * H1 * 4, stream);
  hipMemsetAsync(den1, 0, (size_t)n * H1 * 4, stream);
  hipMemsetAsync(m2,   0, (size_t)n * 4, stream);
  hipMemsetAsync(den2, 0, (size_t)n * 4, stream);

  const int Etot = E + n;

  // ---- conv1 (H=4, C=64) ----
  {
    const int tiles = (n >> 4) * (HC1 >> 6);     // 16x64 strips
    gemm_wmma_f32_x4<<<(tiles + 3) / 4, 128, 0, stream>>>(x, W1, hfeat1, n, EMB, HC1);

    const int pairs = n * H1;
    escore_kernel<H1><<<(pairs + 7) / 8, 256, 0, stream>>>(hfeat1, as1, ad1, esrc1, edst1, n);

    const int eh = Etot * H1;
    edge_max_kernel<H1><<<(eh + 255) / 256, 256, 0, stream>>>(srcI, dstI, esrc1, edst1, m1, E, n);
    edge_denom_kernel<H1><<<(eh + 255) / 256, 256, 0, stream>>>(srcI, dstI, esrc1, edst1, m1, den1, wexp, E, n);

    const long long work = (long long)Etot * (HC1 >> 2);
    edge_agg_kernel<H1><<<(unsigned)((work + 255) / 256), 256, 0, stream>>>(
        srcI, dstI, wexp, den1, hfeat1, out1, E, n);

    const int tot = n * HC1;
    bias_relu_kernel<<<(tot + 255) / 256, 256, 0, stream>>>(out1, b1, tot);
  }

  // ---- conv2 (H=1, C=64) ----
  {
    const int tiles = (n >> 4) * (OUTC >> 6);
    gemm_wmma_f32_x4<<<(tiles + 3) / 4, 128, 0, stream>>>(out1, W2, hfeat2, n, HC1, OUTC);

    escore_kernel<1><<<(n + 7) / 8, 256, 0, stream>>>(hfeat2, as2, ad2, esrc2, edst2, n);

    edge_max_kernel<1><<<(Etot + 255) / 256, 256, 0, stream>>>(srcI, dstI, esrc2, edst2, m2, E, n);
    edge_denom_kernel<1><<<(Etot + 255) / 256, 256, 0, stream>>>(srcI, dstI, esrc2, edst2, m2, den2, wexp, E, n);

    const long long work = (long long)Etot * (OUTC >> 2);
    edge_agg_kernel<1><<<(unsigned)((work + 255) / 256), 256, 0, stream>>>(
        srcI, dstI, wexp, den2, hfeat2, out2, E, n);

    bias_norm_kernel<<<(n + 7) / 8, 256, 0, stream>>>(out2, b2, (float*)d_out, n);
  }
}